// RestorationLoss_76922864271390
// MI455X (gfx1250) — compile-verified
//
#include <hip/hip_runtime.h>
#include <hip/hip_bf16.h>
#include <math.h>

typedef float v2f __attribute__((ext_vector_type(2)));
typedef float v8f __attribute__((ext_vector_type(8)));

#define IMG_H   512
#define IMG_W   512
#define PLANES  48            // 16 * 3
#define TILES_X 16            // 512 / 32
#define TILES_Y 16
#define NBLK    (TILES_X * TILES_Y * PLANES)   // 12288
#define NPIX    (16.0 * 3.0 * 512.0 * 512.0)

// Banded Gaussian weight: g[t] for t in [0,10], else 0. gs = normalizer.
__device__ __forceinline__ float bandw(int t, float gs) {
    if (t < 0 || t > 10) return 0.0f;
    float d = (float)(t - 5);
    return expf(-d * d / 4.5f) / gs;
}

__global__ __launch_bounds__(128) void ssim_tile_kernel(
    const float* __restrict__ img1,
    const float* __restrict__ img2,
    float* __restrict__ partials)
{
    const int lane = threadIdx.x & 31;
    const int wv   = threadIdx.x >> 5;   // wave id 0..3
    const int waveRow = wv & 1;          // 2x2 wave tiles -> 32x32 block tile
    const int waveCol = wv >> 1;

    const int row0 = blockIdx.y * 32 + waveRow * 16;  // output tile origin
    const int col0 = blockIdx.x * 32 + waveCol * 16;
    const size_t planeBase = (size_t)blockIdx.z * (IMG_H * IMG_W);

    // Per-wave horizontal-conv staging: 5 fields x 28 rows (26 + K-pad) x 16 cols,
    // pitch 17 floats to avoid LDS bank conflicts.
    __shared__ float T[4][5][28][17];
    __shared__ float wpart[4][2];

    // ---- Gaussian weights (match reference: normalized 1D, fp32) ----
    const float e1 = expf(-1.0f  / 4.5f);
    const float e2 = expf(-4.0f  / 4.5f);
    const float e3 = expf(-9.0f  / 4.5f);
    const float e4 = expf(-16.0f / 4.5f);
    const float e5 = expf(-25.0f / 4.5f);
    const float gs = 1.0f + 2.0f * (e1 + e2 + e3 + e4 + e5);
    const float g0 = e5 / gs, g1 = e4 / gs, g2 = e3 / gs, g3 = e2 / gs,
                g4 = e1 / gs, g5 = 1.0f / gs;
    const float g[11] = { g0, g1, g2, g3, g4, g5, g4, g3, g2, g1, g0 };

    float mse = 0.0f;

    // ================= Stage A: load slab + horizontal conv (VALU) ==========
    if (lane < 26) {
        const int imgRow = row0 - 5 + lane;
        const bool rok = (unsigned)imgRow < (unsigned)IMG_H;
        float x[26], y[26];
        #pragma unroll
        for (int i = 0; i < 26; ++i) {
            const int icol = col0 - 5 + i;
            float xv = 0.0f, yv = 0.0f;
            if (rok && (unsigned)icol < (unsigned)IMG_W) {
                const size_t idx = planeBase + (size_t)imgRow * IMG_W + (size_t)icol;
                xv = img1[idx];
                yv = img2[idx];
            }
            x[i] = xv; y[i] = yv;
        }

        // MSE over the 16x16 center of the slab (each output pixel exactly once)
        if (lane >= 5 && lane < 21) {
            #pragma unroll
            for (int i = 5; i < 21; ++i) {
                const float d = x[i] - y[i];
                mse = fmaf(d, d, mse);
            }
        }

        // 11-tap horizontal conv, 5 fields, 16 output columns
        #pragma unroll
        for (int c = 0; c < 16; ++c) {
            float hx = 0.f, hy = 0.f, hxx = 0.f, hyy = 0.f, hxy = 0.f;
            #pragma unroll
            for (int k = 0; k < 11; ++k) {
                const float w  = g[k];
                const float xv = x[c + k];
                const float yv = y[c + k];
                hx  = fmaf(w, xv, hx);
                hy  = fmaf(w, yv, hy);
                hxx = fmaf(w * xv, xv, hxx);
                hyy = fmaf(w * yv, yv, hyy);
                hxy = fmaf(w * xv, yv, hxy);
            }
            T[wv][0][lane][c] = hx;
            T[wv][1][lane][c] = hy;
            T[wv][2][lane][c] = hxx;
            T[wv][3][lane][c] = hyy;
            T[wv][4][lane][c] = hxy;
        }
    } else if (lane < 28) {
        // K-padding rows: keep finite (multiplied by exact-zero band weights)
        #pragma unroll
        for (int f = 0; f < 5; ++f)
            #pragma unroll
            for (int c = 0; c < 16; ++c)
                T[wv][f][lane][c] = 0.0f;
    }
    __syncthreads();

    // ================= Stage B: vertical conv via WMMA f32 16x16x4 ==========
    // A[m][j] = g[j - m] (16x28 band), per ISA A-layout:
    //   lanes 0-15: VGPR0 = K=4s+0, VGPR1 = K=4s+1 (row M = lane)
    //   lanes16-31: VGPR0 = K=4s+2, VGPR1 = K=4s+3 (row M = lane-16)
    const int m16 = lane & 15;
    const int khalf = (lane < 16) ? 0 : 2;
    v2f a[7];
    #pragma unroll
    for (int s = 0; s < 7; ++s) {
        const int k0 = 4 * s + khalf;
        a[s].x = bandw(k0 - m16,     gs);
        a[s].y = bandw(k0 + 1 - m16, gs);
    }

    v8f C[5];
    #pragma unroll
    for (int f = 0; f < 5; ++f) {
        v8f acc = {0.f, 0.f, 0.f, 0.f, 0.f, 0.f, 0.f, 0.f};
        #pragma unroll
        for (int s = 0; s < 7; ++s) {
            const int k0 = 4 * s + khalf;
            v2f b;
            // B-layout (row-striped across lanes, K split across lane halves)
            b.x = T[wv][f][k0][m16];
            b.y = T[wv][f][k0 + 1][m16];
            acc = __builtin_amdgcn_wmma_f32_16x16x4_f32(
                false, a[s], false, b, (short)0, acc, false, false);
        }
        C[f] = acc;
    }

    // ================= Pointwise SSIM on accumulators (layout-shared) =======
    const float K1 = 1.0e-4f;   // 0.01^2
    const float K2 = 9.0e-4f;   // 0.03^2
    float ssim = 0.0f;
    #pragma unroll
    for (int v = 0; v < 8; ++v) {
        const float mu1 = C[0][v], mu2 = C[1][v];
        const float mu1s = mu1 * mu1;
        const float mu2s = mu2 * mu2;
        const float m12  = mu1 * mu2;
        const float s1  = C[2][v] - mu1s;
        const float s2  = C[3][v] - mu2s;
        const float s12 = C[4][v] - m12;
        const float num = (2.0f * m12 + K1) * (2.0f * s12 + K2);
        const float den = (mu1s + mu2s + K1) * (s1 + s2 + K2);
        ssim += num / den;
    }

    // ================= Reductions ==========================================
    #pragma unroll
    for (int off = 16; off >= 1; off >>= 1) {
        ssim += __shfl_xor(ssim, off, 32);
        mse  += __shfl_xor(mse,  off, 32);
    }
    if (lane == 0) { wpart[wv][0] = ssim; wpart[wv][1] = mse; }
    __syncthreads();
    if (threadIdx.x == 0) {
        const float S = wpart[0][0] + wpart[1][0] + wpart[2][0] + wpart[3][0];
        const float M = wpart[0][1] + wpart[1][1] + wpart[2][1] + wpart[3][1];
        const int bid = (blockIdx.z * gridDim.y + blockIdx.y) * gridDim.x + blockIdx.x;
        partials[2 * bid]     = S;
        partials[2 * bid + 1] = M;
    }
}

// Deterministic single-block final reduction (fp64 accumulate).
__global__ __launch_bounds__(256) void ssim_finalize_kernel(
    const float* __restrict__ partials, int nblk, float* __restrict__ out)
{
    __shared__ double sd[256];
    __shared__ double md[256];
    const int t = threadIdx.x;
    double s = 0.0, m = 0.0;
    for (int i = t; i < nblk; i += 256) {
        s += (double)partials[2 * i];
        m += (double)partials[2 * i + 1];
    }
    sd[t] = s; md[t] = m;
    __syncthreads();
    for (int off = 128; off >= 1; off >>= 1) {
        if (t < off) { sd[t] += sd[t + off]; md[t] += md[t + off]; }
        __syncthreads();
    }
    if (t == 0) {
        const double loss = (1.0 - sd[0] / NPIX) + md[0] / NPIX;
        out[0] = (float)loss;
    }
}

extern "C" void kernel_launch(void* const* d_in, const int* in_sizes, int n_in,
                              void* d_out, int out_size, void* d_ws, size_t ws_size,
                              hipStream_t stream) {
    const float* r_low  = (const float*)d_in[0];
    const float* r_high = (const float*)d_in[1];
    float* out      = (float*)d_out;
    float* partials = (float*)d_ws;    // NBLK * 2 floats ~= 98 KB

    dim3 grid(TILES_X, TILES_Y, PLANES);
    dim3 block(128);
    ssim_tile_kernel<<<grid, block, 0, stream>>>(r_low, r_high, partials);
    ssim_finalize_kernel<<<1, 256, 0, stream>>>(partials, NBLK, out);
}